// GraphAttention_50105088475801
// MI455X (gfx1250) — compile-verified
//
#include <hip/hip_runtime.h>

typedef float v2f __attribute__((ext_vector_type(2)));
typedef float v8f __attribute__((ext_vector_type(8)));

#define N_NODES 4096
#define F_IN    512
#define H_HEADS 8
#define C_OUT   128
#define ALPHA   0.2f
#define MASKV   (-1.0e10f)

// ---------------------------------------------------------------------------
// Kernel 1: per-head projection  feats[h] = X @ W[h]   ([N,F] x [F,C])
// One wave32 computes one 16x16 output tile with V_WMMA_F32_16X16X4_F32,
// marching K in steps of 4 (128 WMMAs per tile, unrolled x8 so the clause
// scheduler can keep ~20 loads in flight behind the serial WMMA chain).
// ---------------------------------------------------------------------------
__global__ __launch_bounds__(256) void gat_proj(const float* __restrict__ X,
                                                const float* __restrict__ W,
                                                float* __restrict__ feats) {
  const int lane = threadIdx.x & 31;
  const int wave = threadIdx.x >> 5;
  const int hi   = lane >> 4;      // half-wave select (K pairing per ISA layout)
  const int lo   = lane & 15;      // M for A-operand, N for B-operand
  const int rowTile = blockIdx.x * 8 + wave;   // 0..255
  const int cTile   = blockIdx.y;              // 0..7
  const int h       = blockIdx.z;              // 0..7
  const int row0 = rowTile * 16;
  const int c0   = cTile * 16;

  const float* Wh   = W + (size_t)h * F_IN * C_OUT;
  const float* xrow = X + (size_t)(row0 + lo) * F_IN;

  v8f acc = {0.f, 0.f, 0.f, 0.f, 0.f, 0.f, 0.f, 0.f};

#pragma unroll 8
  for (int k = 0; k < F_IN; k += 4) {
    v2f a, b;
    // A (16x4 f32): lanes 0-15 hold K=k,k+1 ; lanes 16-31 hold K=k+2,k+3
    a.x = xrow[k + 2 * hi];
    a.y = xrow[k + 2 * hi + 1];
    // B (4x16 f32): rows of W[h] striped the same way across half-waves
    b.x = Wh[(size_t)(k + 2 * hi) * C_OUT + c0 + lo];
    b.y = Wh[(size_t)(k + 2 * hi + 1) * C_OUT + c0 + lo];
    acc = __builtin_amdgcn_wmma_f32_16x16x4_f32(false, a, false, b,
                                                (short)0, acc, false, false);
  }

  float* outp = feats + ((size_t)h * N_NODES + row0) * C_OUT + c0;
#pragma unroll
  for (int r = 0; r < 8; ++r) {
    const int m = r + 8 * hi;      // C/D layout: VGPR r -> row r (+8 for hi half)
    outp[(size_t)m * C_OUT + lo] = acc[r];
  }
}

// ---------------------------------------------------------------------------
// Kernel 2: attention score pre-reductions
//   s_self[h,n]  = feats[h,n,:] . a_self[h]
//   s_neigh[h,n] = feats[h,n,:] . a_neigh[h]
// One wave per (h,n); 4 elems/lane then shfl_xor tree reduction.
// ---------------------------------------------------------------------------
__global__ __launch_bounds__(256) void gat_scores(const float* __restrict__ feats,
                                                  const float* __restrict__ a_self,
                                                  const float* __restrict__ a_neigh,
                                                  float* __restrict__ s_self,
                                                  float* __restrict__ s_neigh) {
  const int lane = threadIdx.x & 31;
  const int wave = threadIdx.x >> 5;
  const int idx  = blockIdx.x * 8 + wave;      // 0 .. H*N-1  (= h*N + n)
  const int h    = idx >> 12;                  // /4096

  const float* f  = feats + (size_t)idx * C_OUT;
  const float* as = a_self + h * C_OUT;
  const float* an = a_neigh + h * C_OUT;

  float ss = 0.f, sn = 0.f;
#pragma unroll
  for (int c = lane; c < C_OUT; c += 32) {
    const float v = f[c];
    ss += v * as[c];
    sn += v * an[c];
  }
#pragma unroll
  for (int off = 16; off > 0; off >>= 1) {
    ss += __shfl_xor(ss, off, 32);
    sn += __shfl_xor(sn, off, 32);
  }
  if (lane == 0) {
    s_self[idx]  = ss;
    s_neigh[idx] = sn;
  }
}

// ---------------------------------------------------------------------------
// Kernel 3: fused masked-softmax attention aggregation (flash-style).
// Block = (16-row tile, head). 8 waves each own a 16-wide slice of C=128.
// For each 16-col block: scores -> leaky -> mask -> online softmax in LDS,
// then P(16x16) x V(16x16 slice) accumulated via 4x WMMA_F32_16X16X4_F32.
// global_prefetch of next block's A segment + V tile hides first-touch HBM
// latency through the per-iteration barriers.
// Epilogue: /rowsum + bias, ReLU, head-concat store.
// ---------------------------------------------------------------------------
__global__ __launch_bounds__(256) void gat_attn(const float* __restrict__ A,
                                                const float* __restrict__ feats,
                                                const float* __restrict__ s_self,
                                                const float* __restrict__ s_neigh,
                                                const float* __restrict__ bias,
                                                float* __restrict__ out) {
  __shared__ float sc[16][17];       // masked leaky scores
  __shared__ float pp[16][17];       // exp(score - rowmax)
  __shared__ float m_sh[16];         // running row max
  __shared__ float l_sh[16];         // running row sum
  __shared__ float scale_sh[16];     // exp(m_old - m_new)
  __shared__ float ssf[16];          // s_self for this row tile
  __shared__ int   flag;             // any edge in this block?

  const int tid  = threadIdx.x;
  const int lane = tid & 31;
  const int wave = tid >> 5;
  const int hi   = lane >> 4;
  const int lo   = lane & 15;
  const int ti   = tid >> 4;         // 0..15 score row
  const int tj   = tid & 15;         // 0..15 score col
  const int row0 = blockIdx.x * 16;
  const int h    = blockIdx.y;
  const int c0   = wave * 16;        // this wave's C slice

  const float* fh  = feats + (size_t)h * N_NODES * C_OUT;
  const float* snh = s_neigh + h * N_NODES;

  if (tid < 16) {
    ssf[tid]  = s_self[h * N_NODES + row0 + tid];
    m_sh[tid] = -3.0e38f;
    l_sh[tid] = 0.f;
  }
  v8f acc = {0.f, 0.f, 0.f, 0.f, 0.f, 0.f, 0.f, 0.f};
  __syncthreads();

  for (int jb = 0; jb < N_NODES; jb += 16) {
    // prefetch next column block (A mask segment + V tile) one step ahead
    if (jb + 16 < N_NODES) {
      __builtin_prefetch(&A[(size_t)(row0 + ti) * N_NODES + jb + 16 + tj], 0, 3);
      __builtin_prefetch(&fh[(size_t)(jb + 16 + ti) * C_OUT + c0 + tj], 0, 3);
    }

    if (tid == 0) flag = 0;
    __syncthreads();

    const float av = A[(size_t)(row0 + ti) * N_NODES + jb + tj];
    float e = ssf[ti] + snh[jb + tj];
    e = (e > 0.f) ? e : ALPHA * e;           // LeakyReLU before mask (ref order)
    e += MASKV * (1.f - av);
    sc[ti][tj] = e;
    if (av > 0.f) flag = 1;                  // benign race, same value
    __syncthreads();

    if (flag) {                              // uniform across the block
      if (tid < 16) {
        float mb = -3.0e38f;
#pragma unroll
        for (int j = 0; j < 16; ++j) mb = fmaxf(mb, sc[tid][j]);
        const float mo = m_sh[tid];
        const float mn = fmaxf(mo, mb);
        scale_sh[tid] = __expf(mo - mn);
        m_sh[tid]     = mn;
      }
      __syncthreads();

      pp[ti][tj] = __expf(sc[ti][tj] - m_sh[ti]);   // masked entries underflow to 0
      __syncthreads();

      if (tid < 16) {
        float s = 0.f;
#pragma unroll
        for (int j = 0; j < 16; ++j) s += pp[tid][j];
        l_sh[tid] = l_sh[tid] * scale_sh[tid] + s;
      }

      // rescale accumulator by exp(m_old - m_new) per row
#pragma unroll
      for (int r = 0; r < 8; ++r) acc[r] *= scale_sh[r + 8 * hi];

      // P (16x16) x V (16x16 slice of feats) : 4 K-steps of WMMA f32 16x16x4
      const float* vbase = fh + (size_t)jb * C_OUT + c0;
#pragma unroll
      for (int kk = 0; kk < 4; ++kk) {
        const int k = kk * 4 + 2 * hi;
        v2f a, b;
        a.x = pp[lo][k];
        a.y = pp[lo][k + 1];
        b.x = vbase[(size_t)k * C_OUT + lo];
        b.y = vbase[(size_t)(k + 1) * C_OUT + lo];
        acc = __builtin_amdgcn_wmma_f32_16x16x4_f32(false, a, false, b,
                                                    (short)0, acc, false, false);
      }
    }
    __syncthreads();
  }

  // epilogue: softmax normalize, bias, ReLU, concat heads -> [N, H*C]
  const float bv = bias[h * C_OUT + c0 + lo];
#pragma unroll
  for (int r = 0; r < 8; ++r) {
    const int m = r + 8 * hi;
    float o = acc[r] / l_sh[m] + bv;
    o = fmaxf(o, 0.f);
    out[(size_t)(row0 + m) * (H_HEADS * C_OUT) + h * C_OUT + c0 + lo] = o;
  }
}

// ---------------------------------------------------------------------------
extern "C" void kernel_launch(void* const* d_in, const int* in_sizes, int n_in,
                              void* d_out, int out_size, void* d_ws, size_t ws_size,
                              hipStream_t stream) {
  const float* X       = (const float*)d_in[0];  // [N,F]
  const float* A       = (const float*)d_in[1];  // [N,N]
  const float* W       = (const float*)d_in[2];  // [H,F,C]
  const float* a_self  = (const float*)d_in[3];  // [H,C]
  const float* a_neigh = (const float*)d_in[4];  // [H,C]
  const float* bias    = (const float*)d_in[5];  // [H,C]
  float* out = (float*)d_out;                    // [N, H*C]

  // workspace: feats [H,N,C] f32 (16 MB) + s_self [H,N] + s_neigh [H,N]
  float* feats   = (float*)d_ws;
  float* s_self  = feats + (size_t)H_HEADS * N_NODES * C_OUT;
  float* s_neigh = s_self + (size_t)H_HEADS * N_NODES;

  gat_proj<<<dim3(N_NODES / (16 * 8), C_OUT / 16, H_HEADS), 256, 0, stream>>>(X, W, feats);
  gat_scores<<<dim3((H_HEADS * N_NODES) / 8), 256, 0, stream>>>(feats, a_self, a_neigh,
                                                                s_self, s_neigh);
  gat_attn<<<dim3(N_NODES / 16, H_HEADS), 256, 0, stream>>>(A, feats, s_self, s_neigh,
                                                            bias, out);
}